// EmbedMeanField_32701880991880
// MI455X (gfx1250) — compile-verified
//
#include <hip/hip_runtime.h>
#include <math.h>

// ---------------------------------------------------------------------------
// CDNA5 (gfx1250, wave32) EmbedMeanField.
// - GEMMs: V_WMMA_F32_16X16X4_F32, 64-row blocks (B-frag reuse across 4 M-tiles)
// - A tiles staged via GLOBAL_LOAD_ASYNC_TO_LDS_B128 when available
// - Edge scatter: hardware global_atomic_add_f32 (unsafeAtomicAdd)
// ---------------------------------------------------------------------------

typedef __attribute__((ext_vector_type(2))) float v2f;
typedef __attribute__((ext_vector_type(8))) float v8f;
typedef __attribute__((ext_vector_type(4))) int   v4i;

#ifndef __has_builtin
#define __has_builtin(x) 0
#endif

#if __has_builtin(__builtin_amdgcn_global_load_async_to_lds_b128)
#define HAVE_ASYNC_LDS 1
#else
#define HAVE_ASYNC_LDS 0
#endif

#define AS1 __attribute__((address_space(1)))
#define AS3 __attribute__((address_space(3)))

__device__ __forceinline__ v8f wmma4(v2f a, v2f b, v8f c) {
  return __builtin_amdgcn_wmma_f32_16x16x4_f32(false, a, false, b,
                                               (short)0, c, false, false);
}

// copy 16B global -> LDS (per-lane addresses); both must be 16B aligned
__device__ __forceinline__ void cp16_g2l(const float* __restrict__ g, float* l) {
#if HAVE_ASYNC_LDS
  __builtin_amdgcn_global_load_async_to_lds_b128((AS1 v4i*)(float*)g,
                                                 (AS3 v4i*)l, 0, 0);
#else
  *(float4*)l = *(const float4*)g;
#endif
}

__device__ __forceinline__ void async_wait_all() {
#if HAVE_ASYNC_LDS
#if __has_builtin(__builtin_amdgcn_s_wait_asynccnt)
  __builtin_amdgcn_s_wait_asynccnt(0);
#else
  asm volatile("s_wait_asynccnt 0" ::: "memory");
#endif
#endif
}

#define L_DIM   128
#define TWO_L   256
#define EL_DIM  512   // E_TYPES * L

// --------------------------- tanh(sum(wave_embedding)) ---------------------
__global__ __launch_bounds__(128) void k_wavesum(const float* __restrict__ we,
                                                 int kwave,
                                                 float* __restrict__ tw) {
  int j = threadIdx.x;
  float s = 0.f;
  for (int i = 0; i < kwave; ++i) s += we[(size_t)i * L_DIM + j];
  tw[j] = tanhf(s);
}

// --------------------------- input transform (16-row blocks) ---------------
__global__ __launch_bounds__(256) void k_input(const float* __restrict__ nf,
                                               const float* __restrict__ W,
                                               const float* __restrict__ bias,
                                               const float* __restrict__ all_emb,
                                               const float* __restrict__ tw,
                                               float* __restrict__ cur, int N) {
  const int LDI = 132;  // *4B = 528, 16B aligned rows; bank stride 4
  __shared__ float As[16 * 132];
  const int r0 = blockIdx.x * 16;

  for (int i = threadIdx.x; i < 16 * 32; i += 256) {  // 16 rows x 32 float4
    int row = i >> 5, c4 = (i & 31) << 2;
    int rg = min(r0 + row, N - 1);
    cp16_g2l(nf + (size_t)rg * L_DIM + c4, As + row * LDI + c4);
  }
  async_wait_all();
  __syncthreads();

  const int wave = threadIdx.x >> 5;
  const int lane = threadIdx.x & 31;
  const int half = lane >> 4;
  const int mn   = lane & 15;
  const int n0   = wave * 16;

  v8f acc = {};
  for (int k = 0; k < L_DIM; k += 4) {
    const float* ap = As + mn * LDI + k + 2 * half;
    v2f a; a.x = ap[0]; a.y = ap[1];
    const float* bp = W + (size_t)(k + 2 * half) * L_DIM + n0 + mn;
    v2f b; b.x = bp[0]; b.y = bp[L_DIM];
    acc = wmma4(a, b, acc);
  }

  const float bv = bias[n0 + mn];
#pragma unroll
  for (int r = 0; r < 8; ++r) {
    int row = r0 + r + 8 * half;
    if (row < N) {
      int c = n0 + mn;
      float v = acc[r] + bv + all_emb[(size_t)row * L_DIM + c];
      cur[(size_t)row * TWO_L + c] = tanhf(v);
    }
  }
  for (int i = threadIdx.x; i < 16 * L_DIM; i += 256) {
    int row = r0 + (i >> 7), c = i & 127;
    if (row < N) cur[(size_t)row * TWO_L + L_DIM + c] = tw[c];
  }
}

// --------------------------- conv GEMM (64-row blocks) ---------------------
// out[N,512] = cur[N,256] @ W[256,512] + bias
__global__ __launch_bounds__(256) void k_conv(const float* __restrict__ cur,
                                              const float* __restrict__ W,
                                              const float* __restrict__ bias,
                                              float* __restrict__ out, int N) {
  const int LDA = 260;  // 1040B rows: 16B aligned, bank stride 4
  extern __shared__ float As[];
  const int r0 = blockIdx.x * 64;

  for (int i = threadIdx.x; i < 64 * 64; i += 256) {  // 64 rows x 64 float4
    int row = i >> 6, c4 = (i & 63) << 2;
    int rg = min(r0 + row, N - 1);
    cp16_g2l(cur + (size_t)rg * TWO_L + c4, As + row * LDA + c4);
  }
  async_wait_all();
  __syncthreads();

  const int wave = threadIdx.x >> 5;
  const int lane = threadIdx.x & 31;
  const int half = lane >> 4;
  const int mn   = lane & 15;

  v8f acc[4][4] = {};  // [n-tile][m-tile]; 8 waves * 64 cols = 512
  for (int k = 0; k < TWO_L; k += 4) {
    v2f a[4];
#pragma unroll
    for (int m = 0; m < 4; ++m) {
      const float* ap = As + (m * 16 + mn) * LDA + k + 2 * half;
      a[m].x = ap[0]; a[m].y = ap[1];
    }
#pragma unroll
    for (int t = 0; t < 4; ++t) {
      const float* bp = W + (size_t)(k + 2 * half) * EL_DIM + wave * 64 + t * 16 + mn;
      v2f b; b.x = bp[0]; b.y = bp[EL_DIM];
#pragma unroll
      for (int m = 0; m < 4; ++m) acc[t][m] = wmma4(a[m], b, acc[t][m]);
    }
  }
#pragma unroll
  for (int t = 0; t < 4; ++t) {
    int n0 = wave * 64 + t * 16;
    float bv = bias[n0 + mn];
#pragma unroll
    for (int m = 0; m < 4; ++m) {
#pragma unroll
      for (int r = 0; r < 8; ++r) {
        int row = r0 + m * 16 + r + 8 * half;
        if (row < N) out[(size_t)row * EL_DIM + n0 + mn] = acc[t][m][r] + bv;
      }
    }
  }
}

// --------------------------- zero msg --------------------------------------
__global__ __launch_bounds__(256) void k_zero(float4* __restrict__ p, size_t n4) {
  size_t i = (size_t)blockIdx.x * 256 + threadIdx.x;
  size_t stride = (size_t)gridDim.x * 256;
  float4 z; z.x = 0.f; z.y = 0.f; z.z = 0.f; z.w = 0.f;
  for (; i < n4; i += stride) p[i] = z;
}

// --------------------------- edge scatter ----------------------------------
__global__ __launch_bounds__(256) void k_scatter(const float* __restrict__ conv,
                                                 const int* __restrict__ src,
                                                 const int* __restrict__ dst,
                                                 float* __restrict__ msg,
                                                 int ne, int total) {
  const int gw   = (int)((blockIdx.x * 256 + threadIdx.x) >> 5);
  const int lane = threadIdx.x & 31;
  if (gw >= total) return;
  const int e = gw / ne;
  const int i = gw - e * ne;
  const int s = src[(size_t)e * ne + i];
  const int d = dst[(size_t)e * ne + i];
  const size_t off = (size_t)e * L_DIM + lane * 4;
  const float4 v = *(const float4*)(conv + (size_t)s * EL_DIM + off);
  float* p = msg + (size_t)d * EL_DIM + off;
  unsafeAtomicAdd(p + 0, v.x);   // hardware global_atomic_add_f32
  unsafeAtomicAdd(p + 1, v.y);
  unsafeAtomicAdd(p + 2, v.z);
  unsafeAtomicAdd(p + 3, v.w);
}

// --------------------------- merge GEMM (64-row blocks) --------------------
// cur = tanh( tanh(msg) @ W[512,256] + bias + cur )   (in place on cur)
__global__ __launch_bounds__(256) void k_merge(const float* __restrict__ msg,
                                               const float* __restrict__ W,
                                               const float* __restrict__ bias,
                                               float* __restrict__ cur, int N) {
  const int LDM = 516;  // 2064B rows: 16B aligned, bank stride 4
  extern __shared__ float As[];
  const int r0 = blockIdx.x * 64;

  for (int i = threadIdx.x; i < 64 * 128; i += 256) {  // 64 rows x 128 float4
    int row = i >> 7, c4 = (i & 127) << 2;
    int rg = min(r0 + row, N - 1);
    cp16_g2l(msg + (size_t)rg * EL_DIM + c4, As + row * LDM + c4);
  }
  async_wait_all();
  __syncthreads();
  // tanh in place on the staged tile
  for (int i = threadIdx.x; i < 64 * 128; i += 256) {
    int row = i >> 7, c4 = (i & 127) << 2;
    float4* p = (float4*)(As + row * LDM + c4);
    float4 v = *p;
    v.x = tanhf(v.x); v.y = tanhf(v.y); v.z = tanhf(v.z); v.w = tanhf(v.w);
    *p = v;
  }
  __syncthreads();

  const int wave = threadIdx.x >> 5;
  const int lane = threadIdx.x & 31;
  const int half = lane >> 4;
  const int mn   = lane & 15;

  v8f acc[2][4] = {};  // [n-tile][m-tile]; 8 waves * 32 cols = 256
  for (int k = 0; k < EL_DIM; k += 4) {
    v2f a[4];
#pragma unroll
    for (int m = 0; m < 4; ++m) {
      const float* ap = As + (m * 16 + mn) * LDM + k + 2 * half;
      a[m].x = ap[0]; a[m].y = ap[1];
    }
#pragma unroll
    for (int t = 0; t < 2; ++t) {
      const float* bp = W + (size_t)(k + 2 * half) * TWO_L + wave * 32 + t * 16 + mn;
      v2f b; b.x = bp[0]; b.y = bp[TWO_L];
#pragma unroll
      for (int m = 0; m < 4; ++m) acc[t][m] = wmma4(a[m], b, acc[t][m]);
    }
  }
#pragma unroll
  for (int t = 0; t < 2; ++t) {
    int n0 = wave * 32 + t * 16;
    float bv = bias[n0 + mn];
#pragma unroll
    for (int m = 0; m < 4; ++m) {
#pragma unroll
      for (int r = 0; r < 8; ++r) {
        int row = r0 + m * 16 + r + 8 * half;
        if (row < N) {
          size_t idx = (size_t)row * TWO_L + n0 + mn;
          cur[idx] = tanhf(acc[t][m][r] + bv + cur[idx]);
        }
      }
    }
  }
}

// ---------------------------------------------------------------------------
extern "C" void kernel_launch(void* const* d_in, const int* in_sizes, int n_in,
                              void* d_out, int out_size, void* d_ws, size_t ws_size,
                              hipStream_t stream) {
  const float* node_feat = (const float*)d_in[0];
  const float* all_emb   = (const float*)d_in[1];
  const float* wave_emb  = (const float*)d_in[2];
  const float* w_n2l_w   = (const float*)d_in[3];
  const float* w_n2l_b   = (const float*)d_in[4];
  const float* conv_w    = (const float*)d_in[5];
  const float* conv_b    = (const float*)d_in[6];
  const float* merge_w   = (const float*)d_in[7];
  const float* merge_b   = (const float*)d_in[8];
  const int*   edge_src  = (const int*)d_in[9];
  const int*   edge_dst  = (const int*)d_in[10];

  const int N     = in_sizes[0] / L_DIM;
  const int KWAVE = in_sizes[2] / L_DIM;
  const int NE_T  = in_sizes[9];
  const int ne    = NE_T / 4;
  const int MAXLV = in_sizes[5] / (TWO_L * EL_DIM);

  float* cur  = (float*)d_out;
  float* ws   = (float*)d_ws;
  float* tw   = ws;
  float* conv = ws + 256;
  float* msg  = conv + (size_t)N * EL_DIM;

  const int RB16 = (N + 15) / 16;
  const int RB64 = (N + 63) / 64;
  const int conv_lds  = 64 * 260 * 4;   // 66560 B
  const int merge_lds = 64 * 516 * 4;   // 132096 B

  (void)hipFuncSetAttribute((const void*)k_conv,
                            hipFuncAttributeMaxDynamicSharedMemorySize, conv_lds);
  (void)hipFuncSetAttribute((const void*)k_merge,
                            hipFuncAttributeMaxDynamicSharedMemorySize, merge_lds);

  k_wavesum<<<1, 128, 0, stream>>>(wave_emb, KWAVE, tw);
  k_input<<<RB16, 256, 0, stream>>>(node_feat, w_n2l_w, w_n2l_b, all_emb, tw, cur, N);

  const size_t msg_f4   = (size_t)N * EL_DIM / 4;
  const int    zero_grd = 4096;
  const int    sc_grd   = (NE_T + 7) / 8;

  for (int lv = 0; lv < MAXLV; ++lv) {
    k_conv<<<RB64, 256, conv_lds, stream>>>(cur,
                                            conv_w + (size_t)lv * TWO_L * EL_DIM,
                                            conv_b + (size_t)lv * EL_DIM,
                                            conv, N);
    k_zero<<<zero_grd, 256, 0, stream>>>((float4*)msg, msg_f4);
    k_scatter<<<sc_grd, 256, 0, stream>>>(conv, edge_src, edge_dst, msg, ne, NE_T);
    k_merge<<<RB64, 256, merge_lds, stream>>>(msg,
                                              merge_w + (size_t)lv * EL_DIM * TWO_L,
                                              merge_b + (size_t)lv * TWO_L,
                                              cur, N);
  }
}